// BasisAlignment_69295002354328
// MI455X (gfx1250) — compile-verified
//
#include <hip/hip_runtime.h>
#include <hip/hip_bf16.h>

typedef __attribute__((ext_vector_type(2))) float v2f;
typedef __attribute__((ext_vector_type(4))) float v4f;
typedef __attribute__((ext_vector_type(8))) float v8f;

#define N_ROWS 4096
#define N_COLS 1024
#define BIGV   1.0e12f
#define EPSV   1.0e-12f

// ---------------------------------------------------------------------------
// Kernel 1: row means via V_WMMA_F32_16X16X4_F32, one wave per 16-row tile.
// B is the constant 1/1024 in every element, so the K labels inside an A
// fragment are irrelevant: each lane loads a contiguous float4 and feeds TWO
// WMMAs (xy, then zw). Lanes 0-15 cover cols k..k+3 of row (lane&15); lanes
// 16-31 cover cols k+4..k+7 of the same row -> per iteration the 16x8 block
// [k, k+8) of each row is consumed exactly once.
// C/D layout: VGPR r = row r (lanes 0-15) / row r+8 (lanes 16-31); every
// column holds the identical row mean -> extract from lane 0 and lane 16.
// ---------------------------------------------------------------------------
__global__ __launch_bounds__(256)
void ba_row_means_wmma(const float* __restrict__ f1,
                       const float* __restrict__ f2,
                       float* __restrict__ m1,
                       float* __restrict__ m2) {
    const int lane = threadIdx.x & 31;
    const int wave = (int)(blockIdx.x * 8 + (threadIdx.x >> 5)); // 0..511
    const int mat  = wave >> 8;                                  // 0 -> feats1, 1 -> feats2
    const int tile = wave & 255;
    const int rowBase = tile * 16;

    const float* __restrict__ src = (mat == 0) ? f1 : f2;
    float* __restrict__ dst       = (mat == 0) ? m1 : m2;

    const int row  = rowBase + (lane & 15);
    const int koff = (lane >> 4) * 4; // 0 for lanes 0-15, 4 for lanes 16-31
    const float* __restrict__ p = src + (size_t)row * N_COLS + koff;

    v2f b;
    b.x = 1.0f / (float)N_COLS;
    b.y = 1.0f / (float)N_COLS;

    v8f c = {0.f, 0.f, 0.f, 0.f, 0.f, 0.f, 0.f, 0.f};

#pragma unroll 2
    for (int k = 0; k < N_COLS; k += 8) {
        v4f q = *(const v4f*)(p + k);
        v2f a0; a0.x = q.x; a0.y = q.y;
        v2f a1; a1.x = q.z; a1.y = q.w;
        c = __builtin_amdgcn_wmma_f32_16x16x4_f32(
                false, a0, false, b, (short)0, c, false, false);
        c = __builtin_amdgcn_wmma_f32_16x16x4_f32(
                false, a1, false, b, (short)0, c, false, false);
    }

    if (lane == 0) {
#pragma unroll
        for (int r = 0; r < 8; ++r) dst[rowBase + r] = c[r];
    } else if (lane == 16) {
#pragma unroll
        for (int r = 0; r < 8; ++r) dst[rowBase + 8 + r] = c[r];
    }
}

// ---------------------------------------------------------------------------
// Kernel 2: per-row masked min (positive side) / max (negative side) over m2,
// first-occurrence argmin/argmax semantics (matches jnp.argmin/argmax).
// m2 (16 KB) is staged into LDS with GLOBAL_LOAD_ASYNC_TO_LDS_B128 (ASYNCcnt
// DMA path): 256 lanes x 16 B = 4 KB per instruction, 4 rounds.
// ---------------------------------------------------------------------------
__global__ __launch_bounds__(256)
void ba_neighbor_scan(const float* __restrict__ m1,
                      const float* __restrict__ m2,
                      float* __restrict__ denomOut,
                      float* __restrict__ diffOut,
                      float* __restrict__ validOut) {
    __shared__ __align__(16) float sm2[N_ROWS];

    const int t = threadIdx.x;
    // Low 32 bits of a generic pointer to LDS == LDS byte offset (VDST operand).
    const unsigned int ldsBase = (unsigned int)(unsigned long long)(&sm2[0]);

#pragma unroll
    for (int r = 0; r < 4; ++r) {
        unsigned int       ldsAddr = ldsBase + (unsigned int)(r * 4096 + t * 16);
        unsigned long long gAddr   = (unsigned long long)(m2 + r * 1024 + t * 4);
        asm volatile("global_load_async_to_lds_b128 %0, %1, off"
                     :: "v"(ldsAddr), "v"(gAddr)
                     : "memory");
    }
    asm volatile("s_wait_asynccnt 0x0" ::: "memory");
    __syncthreads();

    const int i = (int)(blockIdx.x * 256 + t);
    const float m1i = m1[i];

    float best_l = BIGV;  int ind_l = 0;
    float best_g = -BIGV; int ind_g = 0;

    for (int j = 0; j < N_ROWS; ++j) {
        const float v   = m1i - sm2[j];
        const float lml = (v <= EPSV)  ?  BIGV : v;
        const float lmg = (v >= -EPSV) ? -BIGV : v;
        if (lml < best_l) { best_l = lml; ind_l = j; }
        if (lmg > best_g) { best_g = lmg; ind_g = j; }
    }

    const bool invalid = (best_l == BIGV) || (best_g == -BIGV);
    const float denom  = best_g + best_l;
    const float l_w    = best_g / denom;
    const float g_w    = best_l / denom;
    // dist_map[i, ind] recomputed exactly as the reference indexes it
    const float grad_l = (m1i - sm2[ind_l]) * l_w;
    const float grad_g = (m1i - sm2[ind_g]) * g_w;

    denomOut[i] = denom;
    diffOut[i]  = grad_l - grad_g;
    validOut[i] = invalid ? 0.0f : 1.0f;
}

// ---------------------------------------------------------------------------
// Kernel 3: single block. Phase A: ||denom||_2 over ALL rows (reference norms
// the unmasked vector). Phase B: masked sum of |w * diff| and valid count.
// ---------------------------------------------------------------------------
__global__ __launch_bounds__(256)
void ba_finalize(const float* __restrict__ denomV,
                 const float* __restrict__ diffV,
                 const float* __restrict__ validV,
                 float* __restrict__ out) {
    __shared__ float red[256];
    const int t = threadIdx.x;

    // Phase A: sum of squares of denom
    float ss = 0.0f;
    for (int i = t; i < N_ROWS; i += 256) { const float d = denomV[i]; ss += d * d; }
    red[t] = ss;
    __syncthreads();
    for (int s = 128; s > 0; s >>= 1) {
        if (t < s) red[t] += red[t + s];
        __syncthreads();
    }
    const float norm = sqrtf(red[0]);
    __syncthreads();

    // Phase B: masked |w * diff| and valid count
    float acc = 0.0f, cnt = 0.0f;
    for (int i = t; i < N_ROWS; i += 256) {
        const float val = validV[i];
        const float w   = denomV[i] / (norm + EPSV);
        acc += val * fabsf(w * diffV[i]);
        cnt += val;
    }
    red[t] = acc;
    __syncthreads();
    for (int s = 128; s > 0; s >>= 1) {
        if (t < s) red[t] += red[t + s];
        __syncthreads();
    }
    const float accTot = red[0];
    __syncthreads();

    red[t] = cnt;
    __syncthreads();
    for (int s = 128; s > 0; s >>= 1) {
        if (t < s) red[t] += red[t + s];
        __syncthreads();
    }

    if (t == 0) out[0] = accTot / fmaxf(red[0], 1.0f);
}

// ---------------------------------------------------------------------------
extern "C" void kernel_launch(void* const* d_in, const int* in_sizes, int n_in,
                              void* d_out, int out_size, void* d_ws, size_t ws_size,
                              hipStream_t stream) {
    const float* f1 = (const float*)d_in[0];
    const float* f2 = (const float*)d_in[1];
    float* out = (float*)d_out;

    float* ws     = (float*)d_ws;
    float* m1     = ws;                  // 4096
    float* m2     = ws + N_ROWS;         // 4096
    float* denomV = ws + 2 * N_ROWS;     // 4096
    float* diffV  = ws + 3 * N_ROWS;     // 4096
    float* validV = ws + 4 * N_ROWS;     // 4096

    // 512 tiles (2 matrices x 256 16-row tiles), 8 waves per 256-thread block
    ba_row_means_wmma<<<64, 256, 0, stream>>>(f1, f2, m1, m2);
    ba_neighbor_scan<<<N_ROWS / 256, 256, 0, stream>>>(m1, m2, denomV, diffV, validV);
    ba_finalize<<<1, 256, 0, stream>>>(denomV, diffV, validV, out);
}